// RGCN_54485955117382
// MI455X (gfx1250) — compile-verified
//
#include <hip/hip_runtime.h>

#define NN 50000
#define NE 600000
#define DD 128
#define NLAYERS 3
#define LDSW 132   // padded LDS row stride (floats): conflict-free ds_load_b64

typedef float v2f __attribute__((ext_vector_type(2)));
typedef float v8f __attribute__((ext_vector_type(8)));

// ---------------------------------------------------------------------------
// Zero a float region (float4 granularity; tail handled)
// ---------------------------------------------------------------------------
__global__ void zero_k(float* __restrict__ p, long long n) {
    long long i4 = ((long long)blockIdx.x * blockDim.x + threadIdx.x) * 4;
    if (i4 + 3 < n) {
        *(float4*)(p + i4) = make_float4(0.f, 0.f, 0.f, 0.f);
    } else {
        for (long long j = i4; j < n; ++j) p[j] = 0.f;
    }
}

// ---------------------------------------------------------------------------
// edge_type[e] = argmax(edge_attr[e, 0:2])  (strict > keeps first-max tie rule)
// ---------------------------------------------------------------------------
__global__ void etype_k(const float* __restrict__ ea, int* __restrict__ et, int E) {
    int e = blockIdx.x * blockDim.x + threadIdx.x;
    if (e < E) et[e] = (ea[2 * e + 1] > ea[2 * e]) ? 1 : 0;
}

// ---------------------------------------------------------------------------
// Scatter: 32 lanes per edge; lane c moves h[src][4c..4c+3] into agg_r[dst].
// unsafeAtomicAdd lowers to global_atomic_add_f32 (HW FP atomic at L2).
// ---------------------------------------------------------------------------
__global__ void scatter_k(const float* __restrict__ h,
                          const int* __restrict__ src, const int* __restrict__ dst,
                          const int* __restrict__ et,
                          float* __restrict__ agg0, float* __restrict__ agg1,
                          float* __restrict__ cnt0, float* __restrict__ cnt1, int E) {
    long long tid = (long long)blockIdx.x * blockDim.x + threadIdx.x;
    int e = (int)(tid >> 5);
    int c = (int)(tid & 31);
    if (e >= E) return;
    int s = src[e], d = dst[e], t = et[e];
    const float4 v = *(const float4*)(h + (size_t)s * DD + c * 4);
    float* agg = t ? agg1 : agg0;
    float* p = agg + (size_t)d * DD + c * 4;
    unsafeAtomicAdd(p + 0, v.x);
    unsafeAtomicAdd(p + 1, v.y);
    unsafeAtomicAdd(p + 2, v.z);
    unsafeAtomicAdd(p + 3, v.w);
    if (c == 0) unsafeAtomicAdd((t ? cnt1 : cnt0) + d, 1.0f);
}

// ---------------------------------------------------------------------------
// cnt[i] -> 1 / max(cnt[i], 1)
// ---------------------------------------------------------------------------
__global__ void inv_k(float* __restrict__ c, int n) {
    int i = blockIdx.x * blockDim.x + threadIdx.x;
    if (i < n) c[i] = 1.0f / fmaxf(c[i], 1.0f);
}

// ---------------------------------------------------------------------------
// Fused RGCN layer GEMM using V_WMMA_F32_16X16X4_F32:
//   out = relu( A0@W0 + (A1*inv1)@W1 + (A2*inv2)@W2 + bias )
// One wave computes a 16-row x 128-col output strip (8 C-tiles of v8f).
// Each weight matrix is staged TRANSPOSED into LDS once per block, so the
// B fragment (K-pair per column) is a single conflict-free ds_load_b64.
//   A frag (16x4): lanes 0-15 hold K=k,k+1 ; lanes 16-31 hold K=k+2,k+3.
//   B frag (4x16): lane%16 = column; lane half selects K row pair.
//   C/D tile:      vgpr v, lane l -> row = v + 8*(l>=16), col = l%16.
// ---------------------------------------------------------------------------
__global__ __launch_bounds__(256) void rgcn_gemm(
    const float* __restrict__ A0, const float* __restrict__ A1, const float* __restrict__ A2,
    const float* __restrict__ inv1, const float* __restrict__ inv2,
    const float* __restrict__ W0, const float* __restrict__ W1, const float* __restrict__ W2,
    const float* __restrict__ bias, float* __restrict__ out, int nrows) {
    __shared__ float sW[DD * LDSW];              // 128x128 transposed, padded (66 KB)

    const int tid = threadIdx.x;
    const int lane = tid & 31;
    const int wave = tid >> 5;
    const int strip = blockIdx.x * 8 + wave;     // 16-row strip, wave-uniform
    const int m0 = strip * 16;
    const bool active = (m0 < nrows);            // wave-uniform: EXEC stays all-1s
    const int hs = lane >> 4;                    // lane half select
    const int lr = lane & 15;

    v8f acc[8];
#pragma unroll
    for (int t = 0; t < 8; ++t) {
        float b = active ? bias[t * 16 + lr] : 0.0f;
#pragma unroll
        for (int v = 0; v < 8; ++v) acc[t][v] = b;
    }

    const float* Am[3] = {A0, A1, A2};
    const float* Wm[3] = {W0, W1, W2};
    const float* Iv[3] = {nullptr, inv1, inv2};
    const int arow = m0 + lr;                    // A-fragment row for this lane

    for (int m = 0; m < 3; ++m) {
        const float* W = Wm[m];
        if (!W) continue;                        // block-uniform (kernel arg)

        __syncthreads();                         // previous pass done with sW
        // Stage W transposed: sW[c*LDSW + k] = W[k*DD + c]. Coalesced b128 reads.
#pragma unroll
        for (int i = tid; i < (DD * DD) / 4; i += 256) {
            const int e = i * 4;
            const int k = e >> 7;                // e / 128
            const int c = e & 127;
            const float4 w = *(const float4*)(W + e);
            sW[(c + 0) * LDSW + k] = w.x;
            sW[(c + 1) * LDSW + k] = w.y;
            sW[(c + 2) * LDSW + k] = w.z;
            sW[(c + 3) * LDSW + k] = w.w;
        }
        __syncthreads();
        if (!active) continue;

        const float* A = Am[m];
        const float sc = Iv[m] ? Iv[m][arow] : 1.0f;
        const float* Ar = A + (size_t)arow * DD;
#pragma unroll 4
        for (int kk = 0; kk < 32; ++kk) {
            const int k = kk * 4 + hs * 2;       // this lane-half's K pair
            v2f a = *(const v2f*)(Ar + k);
            a.x *= sc;
            a.y *= sc;
#pragma unroll
            for (int t = 0; t < 8; ++t) {
                const int col = t * 16 + lr;
                const v2f b = *(const v2f*)(&sW[col * LDSW + k]);   // ds_load_b64
                acc[t] = __builtin_amdgcn_wmma_f32_16x16x4_f32(
                    false, a, false, b, (short)0, acc[t], false, false);
            }
        }
    }

    if (active) {
#pragma unroll
        for (int t = 0; t < 8; ++t) {
            const int col = t * 16 + lr;
#pragma unroll
            for (int v = 0; v < 8; ++v) {
                const int row = m0 + hs * 8 + v;
                out[(size_t)row * DD + col] = fmaxf(acc[t][v], 0.0f);
            }
        }
    }
}

// ---------------------------------------------------------------------------
// Host orchestration
// ---------------------------------------------------------------------------
extern "C" void kernel_launch(void* const* d_in, const int* in_sizes, int n_in,
                              void* d_out, int out_size, void* d_ws, size_t ws_size,
                              hipStream_t stream) {
    const float* x  = (const float*)d_in[0];
    const int*   ei = (const int*)d_in[1];      // [2, E]
    const float* ea = (const float*)d_in[2];    // [E, 2]
    const float* Wf = (const float*)d_in[3];    // [128,128]
    const float* bf = (const float*)d_in[4];    // [128]
    const float* cw = (const float*)d_in[5];    // [3,2,128,128]
    const float* cr = (const float*)d_in[6];    // [3,128,128]
    const float* cb = (const float*)d_in[7];    // [3,128]
    const int* src = ei;
    const int* dst = ei + NE;

    // workspace layout (floats): agg0 | agg1 | cnt0 | cnt1 | hA | hB | etype(int)
    float* agg0 = (float*)d_ws;
    float* agg1 = agg0 + (size_t)NN * DD;
    float* cnt0 = agg1 + (size_t)NN * DD;
    float* cnt1 = cnt0 + NN;
    float* hA   = cnt1 + NN;
    float* hB   = hA + (size_t)NN * DD;
    int*   et   = (int*)(hB + (size_t)NN * DD);

    const int gemm_blocks = (NN / 16 + 7) / 8;  // 8 waves (strips) per block

    // edge types
    etype_k<<<(NE + 255) / 256, 256, 0, stream>>>(ea, et, NE);

    // input projection: hA = relu(x @ Wf + bf)
    rgcn_gemm<<<gemm_blocks, 256, 0, stream>>>(x, nullptr, nullptr, nullptr, nullptr,
                                               Wf, nullptr, nullptr, bf, hA, NN);

    float* bufs[2] = {hA, hB};
    for (int l = 0; l < NLAYERS; ++l) {
        const float* hin = bufs[l & 1];
        float* hout = (l == NLAYERS - 1) ? (float*)d_out : bufs[(l + 1) & 1];

        // zero agg0|agg1|cnt0|cnt1 (contiguous span)
        const long long zn = 2LL * NN * DD + 2LL * NN;
        zero_k<<<(int)((zn / 4 + 255) / 256), 256, 0, stream>>>(agg0, zn);

        // scatter-add per relation
        const long long sthreads = (long long)NE * 32;
        scatter_k<<<(int)((sthreads + 255) / 256), 256, 0, stream>>>(
            hin, src, dst, et, agg0, agg1, cnt0, cnt1, NE);

        // reciprocal degree (cnt0,cnt1 contiguous)
        inv_k<<<(2 * NN + 255) / 256, 256, 0, stream>>>(cnt0, 2 * NN);

        // fused: out = relu(hin@root + mean0@W0 + mean1@W1 + bias)
        const float* root = cr + (size_t)l * DD * DD;
        const float* W0   = cw + ((size_t)l * 2 + 0) * DD * DD;
        const float* W1   = cw + ((size_t)l * 2 + 1) * DD * DD;
        const float* bias = cb + (size_t)l * DD;
        rgcn_gemm<<<gemm_blocks, 256, 0, stream>>>(hin, agg0, agg1, cnt0, cnt1,
                                                   root, W0, W1, bias, hout, NN);
    }
}